// NormalizedCC_84988812853474
// MI455X (gfx1250) — compile-verified
//
#include <hip/hip_runtime.h>
#include <cstdint>

// ---------------------------------------------------------------------------
// Fused 3D NCC (9x9x9 box window, valid) on MI455X (gfx1250).
//   - X/Y box sums as band-matrix GEMMs via V_WMMA_F32_16X16X4_F32
//   - Z box sum as running window with 9-slot LDS ring buffer
//   - async global->LDS staging (ASYNCcnt) + prefetch of next slice
//   - 1 wave / WG, 64KB LDS / WG -> 5 WGs per WGP
// ---------------------------------------------------------------------------

typedef float v2f __attribute__((ext_vector_type(2)));
typedef float v8f __attribute__((ext_vector_type(8)));

#define ZD 160
#define YD 192
#define XD 224
#define OZ 152
#define OY 184
#define OX 216
#define WSZ 9
#define INV_W (1.0f / 729.0f)
#define EPS 1e-5f

__device__ __forceinline__ v8f wmma4(v2f a, v2f b, v8f c) {
  // D(16x16,f32) = A(16x4,f32) * B(4x16,f32) + C
  return __builtin_amdgcn_wmma_f32_16x16x4_f32(false, a, false, b, (short)0, c,
                                               false, false);
}

__device__ __forceinline__ float band9(int n, int k) {
  int d = k - n;
  return (d >= 0 && d < WSZ) ? 1.0f : 0.0f;
}

__global__ void __launch_bounds__(32)
ncc_main(const float* __restrict__ im1, const float* __restrict__ im2,
         float* __restrict__ partials) {
  __shared__ float As[32][33];        // image1 slice tile (24x24 used, padded)
  __shared__ float Bs[32][33];        // image2 slice tile
  __shared__ float Is[5][32][17];     // x-summed intermediate per field
  __shared__ float ring[9 * 5 * 256]; // z ring buffer: 9 slices x 5 fields x 16x16
  __shared__ float red[32];

  const int lane  = threadIdx.x;
  const int tx    = blockIdx.x;       // 14 x-tiles of 16 outputs
  const int ty    = blockIdx.y;       // 12 y-tiles of 16 outputs
  const int bz    = blockIdx.z;       // batch*2 + z-chunk
  const int batch = bz >> 1;
  const int chunk = bz & 1;

  const int xo0 = min(16 * tx, OX - 16); // clamped tile origin (ownership-masked)
  const int yo0 = min(16 * ty, OY - 16);
  const int zo0 = chunk * (OZ / 2);      // 0 or 76 (152 = 2*76 exactly)

  const float* a_img = im1 + (size_t)batch * ZD * YD * XD;
  const float* b_img = im2 + (size_t)batch * ZD * YD * XD;

  for (int i = lane; i < 9 * 5 * 256; i += 32) ring[i] = 0.0f;

  const uint32_t as_base = (uint32_t)(uintptr_t)&As[0][0];
  const uint32_t bs_base = (uint32_t)(uintptr_t)&Bs[0][0];

  const v8f vzero = {0, 0, 0, 0, 0, 0, 0, 0};
  v8f S[5];
#pragma unroll
  for (int f = 0; f < 5; ++f) S[f] = vzero;
  float local = 0.0f;

  const int nhalf = lane & 15;  // N (and M) index within 16
  const int hi    = lane >> 4;  // half-wave select
  const int kh    = hi * 2;     // K offset for A/B operand layout
  const int mofs  = hi * 8;     // M offset for C/D layout

  const int nslices = OZ / 2 + 8; // 84 input slices per z-chunk

  for (int zi = 0; zi < nslices; ++zi) {
    const int z = zo0 + zi;
    const size_t gbase = ((size_t)z * YD + yo0) * XD + xo0;

    // All prior LDS reads (X/Y pass, ring) must retire before the async
    // engine overwrites the slice buffers.
    asm volatile("s_wait_dscnt 0" ::: "memory");

    // Async copy 24x24 region of both images into LDS (36 b32 issues/lane).
#pragma unroll
    for (int i = 0; i < 18; ++i) {
      int idx = i * 32 + lane;        // 0..575 == 24*24
      int row = idx / 24;
      int col = idx - row * 24;
      uint32_t loff = (uint32_t)(row * 33 + col) * 4u;
      uint32_t la = as_base + loff;
      uint32_t lb = bs_base + loff;
      uint64_t ga = (uint64_t)(uintptr_t)(a_img + gbase + (size_t)row * XD + col);
      uint64_t gb = (uint64_t)(uintptr_t)(b_img + gbase + (size_t)row * XD + col);
      asm volatile("global_load_async_to_lds_b32 %0, %1, off" ::"v"(la), "v"(ga)
                   : "memory");
      asm volatile("global_load_async_to_lds_b32 %0, %1, off" ::"v"(lb), "v"(gb)
                   : "memory");
    }

    // Prefetch next slice rows into cache while the async copy completes.
    if (zi + 1 < nslices && lane < 24) {
      __builtin_prefetch(a_img + gbase + (size_t)YD * XD + (size_t)lane * XD, 0, 0);
      __builtin_prefetch(b_img + gbase + (size_t)YD * XD + (size_t)lane * XD, 0, 0);
    }

    asm volatile("s_wait_asynccnt 0" ::: "memory");

    // -------- X pass: OUTx[y][xo] = sum_{k=xo..xo+8} F[y][k]  (B = band) ----
#pragma unroll
    for (int t = 0; t < 2; ++t) {      // two M tiles cover y rows 0..23 (+junk)
      v8f acc[5];
#pragma unroll
      for (int f = 0; f < 5; ++f) acc[f] = vzero;
      const int row = nhalf + 16 * t;  // A-matrix M row for this lane
#pragma unroll
      for (int s = 0; s < 6; ++s) {    // K = 24 input x
        const int k0 = 4 * s + kh;
        float a0 = As[row][k0], a1 = As[row][k0 + 1];
        float b0 = Bs[row][k0], b1 = Bs[row][k0 + 1];
        v2f wb; wb.x = band9(nhalf, k0); wb.y = band9(nhalf, k0 + 1);
        v2f va;  va.x  = a0;      va.y  = a1;
        v2f vb;  vb.x  = b0;      vb.y  = b1;
        v2f vaa; vaa.x = a0 * a0; vaa.y = a1 * a1;
        v2f vbb; vbb.x = b0 * b0; vbb.y = b1 * b1;
        v2f vab; vab.x = a0 * b0; vab.y = a1 * b1;
        acc[0] = wmma4(va,  wb, acc[0]);
        acc[1] = wmma4(vb,  wb, acc[1]);
        acc[2] = wmma4(vaa, wb, acc[2]);
        acc[3] = wmma4(vbb, wb, acc[3]);
        acc[4] = wmma4(vab, wb, acc[4]);
      }
#pragma unroll
      for (int f = 0; f < 5; ++f)
#pragma unroll
        for (int v = 0; v < 8; ++v)
          Is[f][16 * t + mofs + v][nhalf] = acc[f][v];
    }

    // -------- Y pass (A = band) + Z running-window update --------
    const int slot = zi % 9;
#pragma unroll
    for (int f = 0; f < 5; ++f) {
      v8f acc = vzero;
#pragma unroll
      for (int s = 0; s < 6; ++s) {    // K = 24 input y
        const int k0 = 4 * s + kh;
        v2f wa; wa.x = band9(nhalf, k0); wa.y = band9(nhalf, k0 + 1);
        v2f bv; bv.x = Is[f][k0][nhalf]; bv.y = Is[f][k0 + 1][nhalf];
        acc = wmma4(wa, bv, acc);
      }
      float* rb = &ring[(slot * 5 + f) * 256];
#pragma unroll
      for (int v = 0; v < 8; ++v) {
        const int p  = v * 32 + lane;
        const float old = rb[p];
        rb[p] = acc[v];
        S[f][v] += acc[v] - old;       // running 9-slice z sum
      }
    }

    // -------- finalize one output z-slice once 9 slices are in S --------
    if (zi >= 8) {
      const int gx = xo0 + nhalf;
      const bool okx = (gx >= 16 * tx); // ownership mask for clamped tiles
#pragma unroll
      for (int v = 0; v < 8; ++v) {
        const int gy = yo0 + mofs + v;
        if (okx && gy >= 16 * ty) {
          const float s1 = S[0][v], s2 = S[1][v];
          const float s11 = S[2][v], s22 = S[3][v], s12 = S[4][v];
          const float cross = s12 - s1 * s2 * INV_W;
          const float var1  = s11 - s1 * s1 * INV_W;
          const float var2  = s22 - s2 * s2 * INV_W;
          local += (cross * cross) / (var1 * var2 + EPS);
        }
      }
    }
  }

  red[lane] = local;
  __syncthreads();
  if (lane == 0) {
    float t = 0.0f;
    for (int i = 0; i < 32; ++i) t += red[i]; // fixed order: deterministic
    partials[((size_t)bz * gridDim.y + ty) * gridDim.x + tx] = t;
  }
}

__global__ void ncc_finalize(const float* __restrict__ partials, int n,
                             float* __restrict__ out, float invN) {
  if (threadIdx.x == 0) {
    float s = 0.0f;
    for (int i = 0; i < n; ++i) s += partials[i]; // fixed order
    out[0] = s * invN;
  }
}

extern "C" void kernel_launch(void* const* d_in, const int* in_sizes, int n_in,
                              void* d_out, int out_size, void* d_ws,
                              size_t ws_size, hipStream_t stream) {
  const float* im1 = (const float*)d_in[0];
  const float* im2 = (const float*)d_in[1];
  float* out = (float*)d_out;
  float* partials = (float*)d_ws;

  dim3 grid(14, 12, 4); // x-tiles, y-tiles, batch*zchunk
  dim3 block(32);       // one wave (wave32) per workgroup
  ncc_main<<<grid, block, 0, stream>>>(im1, im2, partials);

  const int npart = 14 * 12 * 4;
  const float invN = 1.0f / (2.0f * OZ * OY * OX); // 12,082,176 outputs
  ncc_finalize<<<1, 32, 0, stream>>>(partials, npart, out, invN);
}